// Model_78245714198551
// MI455X (gfx1250) — compile-verified
//
#include <hip/hip_runtime.h>

// ---------------------------------------------------------------------------
// GCN layer: h = (segment_sum(h_src[edge_src], edge_dst) + h_src[:n_dst])
//              / (deg + 1) + b,  where h_src = x @ W
// Phase 1: f32 WMMA GEMM (v_wmma_f32_16x16x4_f32). W is staged in LDS
//          pre-swizzled into K-pairs so each lane's B fragment is one
//          ds_load_b64 straight into the even-aligned v2f WMMA operand
//          (no v_mov shuffles between DS returns and WMMAs).
// Phase 2: per-(edge,feature) global_atomic_add_f32 into L2-resident accum.
// Phase 3: elementwise finalize.
// ---------------------------------------------------------------------------

#define IN_F  128
#define OUT_F 64

typedef float v2f __attribute__((ext_vector_type(2)));
typedef float v8f __attribute__((ext_vector_type(8)));

// ---------------- Phase 0: zero accumulators ----------------
__global__ void zero_f32(float* __restrict__ p, int n) {
    int i = blockIdx.x * 256 + threadIdx.x;
    if (i < n) p[i] = 0.0f;
}

// ---------------- Phase 1: h_src = x @ W via WMMA f32 16x16x4 ----------------
// Each wave: one 16-row M tile, full N=64 (4 accumulators of 16x16).
// Block = 256 threads = 8 waves = 128 rows.
__global__ __launch_bounds__(256)
void gemm_xw(const float* __restrict__ x, const float* __restrict__ W,
             float* __restrict__ h, int M) {
    // K-pair swizzled W: Wlds[kp * 64 + n] = { W[2kp][n], W[2kp+1][n] }.
    __shared__ float2 Wlds[(IN_F / 2) * OUT_F];   // 32 KB

    // Cooperative stage + swizzle (coalesced: for fixed kp, n sweeps 0..63).
    for (int i = threadIdx.x; i < (IN_F / 2) * OUT_F; i += 256) {
        int kp = i >> 6;       // k-pair index 0..63
        int n  = i & 63;       // output column
        float2 p;
        p.x = W[(2 * kp + 0) * OUT_F + n];
        p.y = W[(2 * kp + 1) * OUT_F + n];
        Wlds[i] = p;
    }
    __syncthreads();

    const int wave = threadIdx.x >> 5;
    const int lane = threadIdx.x & 31;
    const int half = lane >> 4;      // lane-half selects K offset (ISA A/B layout)
    const int lrow = lane & 15;      // M index for A-frag, N index for B-frag

    const int tile = blockIdx.x * 8 + wave;      // 16-row tile id (uniform per wave)
    if (tile * 16 >= M) return;

    const float* xrow = x + (size_t)(tile * 16 + lrow) * IN_F;
    const int kofs = 2 * half;

    v8f acc[4];
    #pragma unroll
    for (int t = 0; t < 4; ++t) acc[t] = (v8f){};

    for (int k = 0; k < IN_F; k += 4) {
        // A fragment (16x4 f32): lanes 0-15 hold K=k+0,k+1; lanes 16-31 K=k+2,k+3.
        float2 a2 = *(const float2*)(xrow + k + kofs);   // 8B-aligned b64 load
        v2f A; A.x = a2.x; A.y = a2.y;

        // B fragments: one contiguous float2 (ds_load_b64) per n-tile.
        const float2* wrow = Wlds + (size_t)(k / 2 + half) * OUT_F + lrow;

        #pragma unroll
        for (int t = 0; t < 4; ++t) {
            float2 bf = wrow[t * 16];
            v2f B; B.x = bf.x; B.y = bf.y;
            acc[t] = __builtin_amdgcn_wmma_f32_16x16x4_f32(
                /*neg_a=*/false, A, /*neg_b=*/false, B,
                /*c_mod=*/(short)0, acc[t],
                /*reuse_a=*/false, /*reuse_b=*/false);
        }
    }

    // C/D layout: lanes 0-15 rows 0..7 (VGPR index), lanes 16-31 rows 8..15.
    float* out = h + (size_t)tile * 16 * OUT_F;
    #pragma unroll
    for (int v = 0; v < 8; ++v) {
        const int row = 8 * half + v;
        #pragma unroll
        for (int t = 0; t < 4; ++t)
            out[row * OUT_F + t * 16 + lrow] = acc[t][v];
    }
}

// ---------------- Phase 2: edge scatter-add ----------------
// 64 threads per edge: one atomic f32 add per feature. h_src (51 MB) and
// accum (13 MB) are L2-resident (192 MB L2), so atomics resolve in L2.
__global__ __launch_bounds__(256)
void edge_agg(const int* __restrict__ edge_src, const int* __restrict__ edge_dst,
              const float* __restrict__ h_src, float* __restrict__ accum,
              float* __restrict__ deg, int n_edges) {
    int gid = blockIdx.x * 256 + threadIdx.x;
    int e = gid >> 6;          // edge index
    int f = gid & 63;          // feature index
    if (e >= n_edges) return;
    int s = edge_src[e];
    int d = edge_dst[e];
    float v = h_src[(size_t)s * OUT_F + f];
    unsafeAtomicAdd(&accum[(size_t)d * OUT_F + f], v);   // global_atomic_add_f32
    if (f == 0) unsafeAtomicAdd(&deg[d], 1.0f);
}

// ---------------- Phase 3: finalize ----------------
__global__ __launch_bounds__(256)
void finalize(const float* __restrict__ accum, const float* __restrict__ h_src,
              const float* __restrict__ deg, const float* __restrict__ b,
              float* __restrict__ out, int n_dst) {
    int gid = blockIdx.x * 256 + threadIdx.x;
    if (gid >= n_dst * OUT_F) return;
    int i = gid >> 6;          // dst node
    int f = gid & 63;          // feature
    float s = accum[gid] + h_src[gid];          // h_src[:n_dst] is a prefix
    out[gid] = s / (deg[i] + 1.0f) + b[f];
}

// ---------------------------------------------------------------------------
extern "C" void kernel_launch(void* const* d_in, const int* in_sizes, int n_in,
                              void* d_out, int out_size, void* d_ws, size_t ws_size,
                              hipStream_t stream) {
    const float* x    = (const float*)d_in[0];
    const int*   esrc = (const int*)  d_in[1];
    const int*   edst = (const int*)  d_in[2];
    const float* W    = (const float*)d_in[3];
    const float* b    = (const float*)d_in[4];
    // d_in[5] is n_dst on device; derive it from out_size instead (no device read).

    const int n_src   = in_sizes[0] / IN_F;   // 200000
    const int n_edges = in_sizes[1];          // 800000
    const int n_dst   = out_size / OUT_F;     // 50000

    float* h_src = (float*)d_ws;                                   // n_src*64
    float* accum = h_src + (size_t)n_src * OUT_F;                  // n_dst*64
    float* deg   = accum + (size_t)n_dst * OUT_F;                  // n_dst

    // Zero accum + deg (contiguous region) every launch: graph-replay safe.
    {
        int nz = n_dst * (OUT_F + 1);
        zero_f32<<<(nz + 255) / 256, 256, 0, stream>>>(accum, nz);
    }

    // GEMM: 8 wave-tiles (16 rows each) per block.
    {
        int mtiles = n_src / 16;              // 12500 (n_src % 16 == 0)
        int blocks = (mtiles + 7) / 8;        // 1563
        gemm_xw<<<blocks, 256, 0, stream>>>(x, W, h_src, n_src);
    }

    // Edge aggregation: 64 threads per edge.
    {
        long long total = (long long)n_edges * OUT_F;              // 51.2M
        int blocks = (int)((total + 255) / 256);
        edge_agg<<<blocks, 256, 0, stream>>>(esrc, edst, h_src, accum, deg, n_edges);
    }

    // Finalize.
    {
        int total = n_dst * OUT_F;
        finalize<<<(total + 255) / 256, 256, 0, stream>>>(accum, h_src, deg, b,
                                                          (float*)d_out, n_dst);
    }
}